// BinsChamferLoss_39324720562919
// MI455X (gfx1250) — compile-verified
//
#include <hip/hip_runtime.h>
#include <hip/hip_bf16.h>
#include <cstdint>

// ---- problem constants (from reference) ----
#define NB 8
#define PB 256
#define HT 240
#define WD 320
#define TT (HT * WD)                 // 76800 points per batch
#define BLOCK 512                    // 16 wave32s -> covers all 16 p-tiles
#define WAVES 16
#define CHUNK_TILES 96               // t-tiles per workgroup
#define CHUNK_T (CHUNK_TILES * 16)   // 1536 points per workgroup
#define CHUNKS (TT / CHUNK_T)        // 50 chunks per batch
#define BIGG 1.0e9f                  // masked-out gt sentinel -> d2 ~ 1e18 (finite)
#define FMAXBITS 0x7F7FFFFFu         // bits of FLT_MAX

typedef __attribute__((ext_vector_type(2))) float v2f;  // A/B frag: 16x4 / 4x16 f32
typedef __attribute__((ext_vector_type(8))) float v8f;  // C/D frag: 16x16 f32

// unsigned min: on non-negative IEEE f32 bit patterns this equals float min,
// and it lowers to v_min_u32 / v_min3_u32 with NO NaN-canonicalize ops.
__device__ __forceinline__ unsigned int umin32(unsigned int a, unsigned int b) {
  return a < b ? a : b;
}

// ---------------- workspace init (poisoned between runs -> must re-init) ----
__global__ void bcl_init_ws(unsigned int* __restrict__ chamx,
                            float* __restrict__ chamy_sum,
                            float* __restrict__ ylen) {
  int i = blockIdx.x * blockDim.x + threadIdx.x;
  if (i < NB * PB) chamx[i] = FMAXBITS;
  if (i < NB) { chamy_sum[i] = 0.0f; ylen[i] = 0.0f; }
}

// ---------------- main chamfer kernel --------------------------------------
// d2[p,t] = (c_p - g_t)^2 = [c^2, -2c, 1] . [1, g, g^2]^T  => K=3 matmul,
// computed 16x16 tiles at a time with V_WMMA_F32_16X16X4_F32.
__global__ __launch_bounds__(BLOCK)
void bcl_main(const float* __restrict__ gt,
              const int* __restrict__ mask,        // depth_mask (integer per harness)
              const float* __restrict__ edges,     // (NB, PB+1)
              unsigned int* __restrict__ chamx_g,  // (NB, PB) f32 bits, global min
              float* __restrict__ chamy_sum_g,     // (NB,)
              float* __restrict__ ylen_g) {        // (NB,)
  __shared__ float        s_g[CHUNK_T];            // masked gt chunk
  __shared__ unsigned int s_cy[CHUNK_T];           // per-t min over p (f32 bits)
  __shared__ float        s_sum, s_cnt;

  const int n     = blockIdx.x / CHUNKS;
  const int chunk = blockIdx.x % CHUNKS;
  const int tid   = threadIdx.x;
  const int lane  = tid & 31;                      // wave32
  const int wave  = tid >> 5;
  const bool hi   = (lane & 16) != 0;              // upper half-wave

  // --- A fragment (per wave, built once): bins p = wave*16 .. wave*16+15 ---
  // lanes 0-15: M=lane, VGPR0=K0, VGPR1=K1 ; lanes 16-31: M=lane-16, K2,K3
  const int   p  = wave * 16 + (lane & 15);
  const float e0 = edges[n * (PB + 1) + p];
  const float e1 = edges[n * (PB + 1) + p + 1];
  const float c  = 0.5f * (e0 + e1);
  v2f afrag;
  afrag.x = hi ? 1.0f : c * c;      // K0 = c^2   | K2 = 1
  afrag.y = hi ? 0.0f : -2.0f * c;  // K1 = -2c   | K3 = 0

  // --- stage masked gt chunk into LDS, init per-t min accumulators ---------
  const long base = (long)n * TT + (long)chunk * CHUNK_T;
  for (int i = tid; i < CHUNK_T; i += BLOCK) {
    const float g = gt[base + i];
    s_g[i]  = mask[base + i] ? g : BIGG;           // invalid points pushed far away
    s_cy[i] = FMAXBITS;
  }
  if (tid == 0) { s_sum = 0.0f; s_cnt = 0.0f; }
  __syncthreads();

  unsigned int cx[8];
#pragma unroll
  for (int r = 0; r < 8; ++r) cx[r] = FMAXBITS;

  // --- sweep t-tiles: one WMMA per 16x16 tile ------------------------------
  for (int t = 0; t < CHUNK_TILES; ++t) {
    const float g = s_g[t * 16 + (lane & 15)];
    v2f bfrag;                                     // B col t = (1, g, g^2, 0)
    bfrag.x = hi ? g * g : 1.0f;                   // K2 | K0
    bfrag.y = hi ? 0.0f  : g;                      // K3 | K1
    v8f cz = {0.f, 0.f, 0.f, 0.f, 0.f, 0.f, 0.f, 0.f};
    v8f d = __builtin_amdgcn_wmma_f32_16x16x4_f32(
        /*neg_a=*/false, afrag, /*neg_b=*/false, bfrag,
        /*c_mod=*/(short)0, cz, /*reuse_a=*/false, /*reuse_b=*/false);

    // all mins in u32 space (d2 >= 0, never NaN) -> v_min_u32/v_min3_u32,
    // no NaN-canonicalize VALU overhead
    unsigned int cy = __float_as_uint(d[0]);
#pragma unroll
    for (int r = 0; r < 8; ++r) {
      const unsigned int du = __float_as_uint(d[r]);
      cx[r] = umin32(cx[r], du);                   // min over t (per p row)
      cy    = umin32(cy, du);                      // min over 8 M values (per t)
    }
    // combine half-waves + all 16 waves for this t
    atomicMin(&s_cy[t * 16 + (lane & 15)], cy);
  }

  // --- cham_x: reduce the 16 t-lanes per (wave, M), fold into global min ---
#pragma unroll
  for (int r = 0; r < 8; ++r) {
    unsigned int v = cx[r];
    v = umin32(v, (unsigned int)__shfl_xor((int)v, 1, 32));
    v = umin32(v, (unsigned int)__shfl_xor((int)v, 2, 32));
    v = umin32(v, (unsigned int)__shfl_xor((int)v, 4, 32));
    v = umin32(v, (unsigned int)__shfl_xor((int)v, 8, 32)); // full min per half
    if ((lane & 15) == 0) {
      const int pp = wave * 16 + (hi ? r + 8 : r);
      atomicMin(&chamx_g[n * PB + pp], v);
    }
  }
  __syncthreads();

  // --- cham_y: masked sum of per-t mins + valid count ----------------------
  float mysum = 0.0f, mycnt = 0.0f;
  for (int i = tid; i < CHUNK_T; i += BLOCK) {
    if (mask[base + i]) {
      mysum += __uint_as_float(s_cy[i]);
      mycnt += 1.0f;
    }
  }
#pragma unroll
  for (int o = 1; o < 32; o <<= 1) {
    mysum += __shfl_xor(mysum, o, 32);
    mycnt += __shfl_xor(mycnt, o, 32);
  }
  if (lane == 0) { atomicAdd(&s_sum, mysum); atomicAdd(&s_cnt, mycnt); }
  __syncthreads();
  if (tid == 0) {
    atomicAdd(&chamy_sum_g[n], s_sum);
    atomicAdd(&ylen_g[n], s_cnt);
  }
}

// ---------------- final scalar combine -------------------------------------
__global__ void bcl_finalize(const unsigned int* __restrict__ chamx,
                             const float* __restrict__ chamy_sum,
                             const float* __restrict__ ylen,
                             float* __restrict__ out) {
  __shared__ float red[PB];
  const int tid = threadIdx.x;
  float total = 0.0f;
  for (int n = 0; n < NB; ++n) {
    red[tid] = __uint_as_float(chamx[n * PB + tid]);
    __syncthreads();
    for (int s = PB / 2; s > 0; s >>= 1) {
      if (tid < s) red[tid] += red[tid + s];
      __syncthreads();
    }
    if (tid == 0) total += red[0] / (float)PB + chamy_sum[n] / ylen[n];
    __syncthreads();
  }
  if (tid == 0) out[0] = total / (float)NB;
}

// ---------------- launcher --------------------------------------------------
extern "C" void kernel_launch(void* const* d_in, const int* in_sizes, int n_in,
                              void* d_out, int out_size, void* d_ws, size_t ws_size,
                              hipStream_t stream) {
  // inputs (setup_inputs order): depth_pred (unused), depth_gt, depth_mask, bin_edges
  const float* gt    = (const float*)d_in[1];
  const int*   mask  = (const int*)d_in[2];
  const float* edges = (const float*)d_in[3];
  float* out = (float*)d_out;

  // workspace layout
  unsigned int* chamx     = (unsigned int*)d_ws;                 // NB*PB u32
  float*        chamy_sum = (float*)(chamx + NB * PB);           // NB f32
  float*        ylen      = chamy_sum + NB;                      // NB f32

  bcl_init_ws<<<(NB * PB + 255) / 256, 256, 0, stream>>>(chamx, chamy_sum, ylen);
  bcl_main<<<NB * CHUNKS, BLOCK, 0, stream>>>(gt, mask, edges,
                                              chamx, chamy_sum, ylen);
  bcl_finalize<<<1, PB, 0, stream>>>(chamx, chamy_sum, ylen, out);
}